// CPUForgetMult_23948737642763
// MI455X (gfx1250) — compile-verified
//
#include <hip/hip_runtime.h>
#include <stdint.h>

// ForgetMult scan: h_t = f_t*x_t + (1-f_t)*h_{t-1}, T sequential, B*H parallel.
// Memory-bound (402 MB @ 23.3 TB/s ~= 17us). Only 2048 waves of parallelism,
// so latency hiding comes from the CDNA5 async global->LDS engine (ASYNCcnt),
// which keeps ~32 B128 transfers (16KB) in flight per wave with zero VGPR cost.

#define LANES        32
#define WAVES        8
#define CH_PER_BLOCK 256   // WAVES * LANES
#define BATCH_T      16    // timesteps per staged batch
#define DEPTH        4     // batches in flight (quad buffer)
// per batch per wave: f + x, each 16t x 32ch x 4B = 2KB = 4 x b128 async ops
// => 8 async ops per batch, 32 outstanding at steady state (< 63 ASYNCcnt max)

__device__ __forceinline__ void issue_tile(const float* __restrict__ gptr,
                                           unsigned lds_base, int slot,
                                           int t0, int BH,
                                           int trow, int cseg, int goff_lane) {
  // One b128 per lane: lane L covers t = t0 + g*4 + (L>>3), channels
  // ch_base + (L&7)*4 .. +3. Coalesced 512B per op, 16B aligned.
#pragma unroll
  for (int g = 0; g < 4; ++g) {
    const int t = t0 + g * 4 + trow;
    const int voff = t * BH * 4 + goff_lane;                  // byte offset vs SADDR
    const unsigned ldsd = lds_base +
        (unsigned)(slot * (BATCH_T * LANES * 4) + (g * 4 + trow) * (LANES * 4) + cseg);
    asm volatile("global_load_async_to_lds_b128 %0, %1, %2"
                 :: "v"(ldsd), "v"(voff), "s"(gptr)
                 : "memory");
  }
}

__device__ __forceinline__ float consume_tile(const float* lf, const float* lx,
                                              float h, float* __restrict__ out,
                                              long long base, int BH) {
#pragma unroll
  for (int tt = 0; tt < BATCH_T; ++tt) {
    const float fv = lf[tt * LANES];   // ds_load_b32, bank-conflict free
    const float xv = lx[tt * LANES];
    h = fmaf(fv, xv, (1.0f - fv) * h);
    __builtin_nontemporal_store(h, out + base + (long long)tt * BH);
  }
  return h;
}

__global__ void __launch_bounds__(CH_PER_BLOCK)
forgetmult_scan_async(const float* __restrict__ f, const float* __restrict__ x,
                      const float* __restrict__ h0, float* __restrict__ out,
                      int nbatch, int BH) {
  // Per-wave private staging: 2 arrays x DEPTH x BATCH_T x LANES floats = 16KB/wave,
  // 128KB per workgroup (fits 2 WGs in the 320KB WGP LDS).
  __shared__ float lds_f[WAVES][DEPTH][BATCH_T][LANES];
  __shared__ float lds_x[WAVES][DEPTH][BATCH_T][LANES];

  const int wave = threadIdx.x >> 5;
  const int lane = threadIdx.x & 31;
  const int ch_base = blockIdx.x * CH_PER_BLOCK + wave * LANES;
  const int ch = ch_base + lane;
  const int trow = lane >> 3;          // 0..3 : t-row within a 4-row group
  const int cseg = (lane & 7) << 4;    // byte offset of this lane's 16B segment
  const int goff_lane = ch_base * 4 + cseg;

  const unsigned fb = (unsigned)(uintptr_t)&lds_f[wave][0][0][0];
  const unsigned xb = (unsigned)(uintptr_t)&lds_x[wave][0][0][0];

  // Prologue: fill the pipeline (DEPTH batches = 32 async ops in flight).
#pragma unroll
  for (int k = 0; k < DEPTH; ++k) {
    issue_tile(f, fb, k, k * BATCH_T, BH, trow, cseg, goff_lane);
    issue_tile(x, xb, k, k * BATCH_T, BH, trow, cseg, goff_lane);
  }

  float h = h0[ch];
  int k = 0;
  // Steady state: 32 outstanding; wait <=24 completes the oldest batch
  // (async loads complete in order).
  for (; k + DEPTH < nbatch; ++k) {
    asm volatile("s_wait_asynccnt 24" ::: "memory");
    const int slot = k & (DEPTH - 1);
    h = consume_tile(&lds_f[wave][slot][0][lane], &lds_x[wave][slot][0][lane],
                     h, out, (long long)k * BATCH_T * BH + ch, BH);
    issue_tile(f, fb, slot, (k + DEPTH) * BATCH_T, BH, trow, cseg, goff_lane);
    issue_tile(x, xb, slot, (k + DEPTH) * BATCH_T, BH, trow, cseg, goff_lane);
  }
  // Drain epilogue: thresholds 24/16/8/0 as outstanding batches shrink 4->1.
  {
    asm volatile("s_wait_asynccnt 24" ::: "memory");
    const int slot = k & (DEPTH - 1);
    h = consume_tile(&lds_f[wave][slot][0][lane], &lds_x[wave][slot][0][lane],
                     h, out, (long long)k * BATCH_T * BH + ch, BH);
    ++k;
  }
  {
    asm volatile("s_wait_asynccnt 16" ::: "memory");
    const int slot = k & (DEPTH - 1);
    h = consume_tile(&lds_f[wave][slot][0][lane], &lds_x[wave][slot][0][lane],
                     h, out, (long long)k * BATCH_T * BH + ch, BH);
    ++k;
  }
  {
    asm volatile("s_wait_asynccnt 8" ::: "memory");
    const int slot = k & (DEPTH - 1);
    h = consume_tile(&lds_f[wave][slot][0][lane], &lds_x[wave][slot][0][lane],
                     h, out, (long long)k * BATCH_T * BH + ch, BH);
    ++k;
  }
  {
    asm volatile("s_wait_asynccnt 0" ::: "memory");
    const int slot = k & (DEPTH - 1);
    h = consume_tile(&lds_f[wave][slot][0][lane], &lds_x[wave][slot][0][lane],
                     h, out, (long long)k * BATCH_T * BH + ch, BH);
  }
}

// Fallback for shapes not matching the tiling assumptions.
__global__ void forgetmult_generic(const float* __restrict__ f,
                                   const float* __restrict__ x,
                                   const float* __restrict__ h0,
                                   float* __restrict__ out,
                                   int T, long long BH) {
  const long long ch = (long long)blockIdx.x * blockDim.x + threadIdx.x;
  if (ch >= BH) return;
  float h = h0[ch];
  for (int t = 0; t < T; ++t) {
    const long long idx = (long long)t * BH + ch;
    const float fv = f[idx];
    const float xv = x[idx];
    h = fmaf(fv, xv, (1.0f - fv) * h);
    out[idx] = h;
  }
}

extern "C" void kernel_launch(void* const* d_in, const int* in_sizes, int n_in,
                              void* d_out, int out_size, void* d_ws, size_t ws_size,
                              hipStream_t stream) {
  (void)n_in; (void)out_size; (void)d_ws; (void)ws_size;
  const float* f  = (const float*)d_in[0];   // [T, B, H] forget gate
  const float* x  = (const float*)d_in[1];   // [T, B, H]
  const float* h0 = (const float*)d_in[2];   // [B, H]
  float* out = (float*)d_out;                // [T, B, H]

  const long long BH = in_sizes[2];
  const long long total = in_sizes[0];
  const int T = (int)(total / BH);

  const bool fast = (BH % CH_PER_BLOCK == 0) && (T % BATCH_T == 0) &&
                    (T >= DEPTH * BATCH_T) && (total * 4 < (1LL << 31));
  if (fast) {
    const int nbatch = T / BATCH_T;
    dim3 grid((unsigned)(BH / CH_PER_BLOCK));
    forgetmult_scan_async<<<grid, CH_PER_BLOCK, 0, stream>>>(
        f, x, h0, out, nbatch, (int)BH);
  } else {
    const int threads = 256;
    const long long blocks = (BH + threads - 1) / threads;
    forgetmult_generic<<<(unsigned)blocks, threads, 0, stream>>>(f, x, h0, out, T, BH);
  }
}